// AnnularPhotonicTransferMatrix_39075612459059
// MI455X (gfx1250) — compile-verified
//
#include <hip/hip_runtime.h>
#include <hip/hip_bf16.h>
#include <math.h>

// ---------------------------------------------------------------------------
// Annular photonic transfer-matrix reflectance sweep, MI455X (gfx1250).
// Roofline: 56 B/point * 2M points = ~117 MB -> ~5 us at 23.3 TB/s, vs
// ~2.5 KFLOP/point of fp32 VALU with ~110 quarter-rate transcendentals ->
// tens of us of VALU. Transcendental/VALU-bound: pin every rcp/rsq/sqrt to a
// single hardware instruction, share the Bessel sub-expressions, keep the
// 2x2 complex chain in its real [[a,ib],[ic,d]] closed form. WMMA does not
// apply (per-lane 2x2 chains; coefficients 1e-7..1e14 force fp32).
// ---------------------------------------------------------------------------

#define PI_2_F         1.57079632679489662f
#define TWO_OVER_PI_F  0.636619772f
#define SQRT_2OPI_F    0.79788456080286536f   // sqrt(2/pi)

// Raw hardware transcendentals (single v_rcp_f32 / v_rsq_f32 / v_sqrt_f32),
// with safe fallbacks if the builtin is unavailable.
__device__ __forceinline__ float hrcp(float x) {
#if __has_builtin(__builtin_amdgcn_rcpf)
    return __builtin_amdgcn_rcpf(x);
#else
    return __fdividef(1.0f, x);
#endif
}
__device__ __forceinline__ float hrsq(float x) {
#if __has_builtin(__builtin_amdgcn_rsqf)
    return __builtin_amdgcn_rsqf(x);
#else
    return __fdividef(1.0f, sqrtf(x));
#endif
}
__device__ __forceinline__ float hsqrt(float x) {
#if __has_builtin(__builtin_amdgcn_sqrtf)
    return __builtin_amdgcn_sqrtf(x);
#else
    return sqrtf(x);
#endif
}

__device__ __forceinline__ float poly4f(float y, float c4, float c3, float c2,
                                        float c1, float c0) {
    float a = c4;
    a = fmaf(a, y, c3);
    a = fmaf(a, y, c2);
    a = fmaf(a, y, c1);
    a = fmaf(a, y, c0);
    return a;
}

__device__ __forceinline__ float poly5f(float y, float c5, float c4, float c3,
                                        float c2, float c1, float c0) {
    float a = c5;
    a = fmaf(a, y, c4);
    a = fmaf(a, y, c3);
    a = fmaf(a, y, c2);
    a = fmaf(a, y, c1);
    a = fmaf(a, y, c0);
    return a;
}

__device__ __forceinline__ float poly6f(float y, float c6, float c5, float c4,
                                        float c3, float c2, float c1, float c0) {
    float a = c6;
    a = fmaf(a, y, c5);
    a = fmaf(a, y, c4);
    a = fmaf(a, y, c3);
    a = fmaf(a, y, c2);
    a = fmaf(a, y, c1);
    a = fmaf(a, y, c0);
    return a;
}

// Numerical-Recipes J0, J1, Y0, Y1 evaluated together for one x > 0.
// Shares y = x*x, 1/x, z = 8/x, z*z, sqrt(2/(pi x)) (via v_rsq), log(x),
// and a single sin/cos pair (x - 3pi/4 == (x - pi/4) - pi/2).
__device__ __forceinline__ void bessel4(float x, float& rj0, float& rj1,
                                        float& ry0, float& ry1) {
    const float y = x * x;

    // ---------------- small-argument rationals (x < 8) ----------------
    // J0
    float n = poly5f(y, -184.9052456f, 77392.33017f, -11214424.18f,
                     651619640.7f, -13362590354.0f, 57568490574.0f);
    float d = poly5f(y, 1.0f, 267.8532712f, 59272.64853f, 9494680.718f,
                     1029532985.0f, 57568490411.0f);
    const float j0s = n * hrcp(d);
    // J1
    n = x * poly5f(y, -30.16036606f, 15704.48260f, -2972611.439f,
                   242396853.1f, -7895059235.0f, 72362614232.0f);
    d = poly5f(y, 1.0f, 376.9991397f, 99447.43394f, 18583304.74f,
               2300535178.0f, 144725228442.0f);
    const float j1s = n * hrcp(d);

    const float logx = __logf(x);
    const float rx = hrcp(x);

    // Y0
    n = poly5f(y, 228.4622733f, -86327.92757f, 10879881.29f,
               -512359803.6f, 7062834065.0f, -2957821389.0f);
    d = poly5f(y, 1.0f, 226.1030244f, 47447.26470f, 7189466.438f,
               745249964.8f, 40076544269.0f);
    const float y0s = fmaf(TWO_OVER_PI_F * j0s, logx, n * hrcp(d));
    // Y1
    n = x * poly5f(y, 8.511937935e4f, -4.237922726e7f, 7.349264551e9f,
                   -5.153438139e11f, 1.275274390e13f, -4.900604943e13f);
    d = poly6f(y, 1.0f, 3.549632885e3f, 1.020426050e6f, 2.245904002e8f,
               3.733650367e10f, 4.244419664e12f, 2.499580570e14f);
    const float y1s = fmaf(TWO_OVER_PI_F, fmaf(j1s, logx, -rx), n * hrcp(d));

    // ---------------- large-argument asymptotics (x >= 8) ----------------
    const float z = 8.0f * rx;
    const float y2 = z * z;
    const float p0c = poly4f(y2, 0.2093887211e-6f, -0.2073370639e-5f,
                             0.2734510407e-4f, -0.1098628627e-2f, 1.0f);
    const float q0c = poly4f(y2, -0.934935152e-7f, 0.7621095161e-6f,
                             -0.6911147651e-5f, 0.1430488765e-3f,
                             -0.1562499995e-1f);
    const float p1c = poly4f(y2, -0.240337019e-6f, 0.2457520174e-5f,
                             -0.3516396496e-4f, 0.183105e-2f, 1.0f);
    const float q1c = poly4f(y2, 0.105787412e-6f, -0.88228987e-6f,
                             0.8449199096e-5f, -0.2002690873e-3f,
                             0.04687499995f);
    const float amp = SQRT_2OPI_F * hrsq(x);   // sqrt(2/(pi*x)) via v_rsq_f32
    // c = cos(x - pi/4), s = sin(x - pi/4);
    // then cos(x-3pi/4) = s, sin(x-3pi/4) = -c.
    const float xx = x - 0.785398164f;
    const float c = __cosf(xx);
    const float s = __sinf(xx);
    const float j0b = amp * fmaf(-z * s, q0c, c * p0c);
    const float y0b = amp * fmaf(z * c, q0c, s * p0c);
    const float j1b = amp * fmaf(z * c, q1c, s * p1c);
    const float y1b = amp * fmaf(z * s, q1c, -c * p1c);

    const bool sm = x < 8.0f;
    rj0 = sm ? j0s : j0b;
    rj1 = sm ? j1s : j1b;
    ry0 = sm ? y0s : y0b;
    ry1 = sm ? y1s : y1b;
}

// Shell transfer matrix for m = 0 has the form [[a, i*b],[i*c, d]] with real
// a,b,c,d (J0' = -J1, Y0' = -Y1). This form is closed under multiplication.
__device__ __forceinline__ void shell_matrix(float k, float p, float invp,
                                             float r0, float r1, float& a,
                                             float& b, float& c, float& d) {
    const float x0 = k * r0;
    const float x1 = k * r1;
    float j0a, j1a, y0a, y1a, j0b, j1b, y0b, y1b;
    bessel4(x0, j0a, j1a, y0a, y1a);
    bessel4(x1, j0b, j1b, y0b, y1b);
    const float pref = PI_2_F * x0;
    a = pref * fmaf(j1a, y0b, -y1a * j0b);
    b = (pref * invp) * fmaf(j0a, y0b, -y0a * j0b);
    c = (p * pref) * fmaf(j1a, y1b, -y1a * j1b);
    d = pref * fmaf(y0a, j1b, -j0a * y1b);
}

__global__ __launch_bounds__(256)
void AnnularPhotonicTransferMatrix_39075612459059_kernel(
    const float* __restrict__ omega,  // [W]
    const float* __restrict__ eps,    // [6, W]
    const float* __restrict__ mu,     // [6, W]
    const float* __restrict__ rho,    // [6, 2] (uniform -> scalar loads)
    float* __restrict__ out,          // [W]
    int W) {
    const int i = blockIdx.x * blockDim.x + threadIdx.x;
    if (i >= W) return;

    // Stream-ahead prefetch (gfx1250 global_prefetch_b8): lines 8 blocks (2048 points) ahead.
    {
        const int pf = i + 256 * 8;
        if (pf < W) {
            __builtin_prefetch(omega + pf, 0, 1);
#pragma unroll
            for (int l = 0; l < 6; ++l) {
                __builtin_prefetch(eps + l * (size_t)W + pf, 0, 1);
                __builtin_prefetch(mu + l * (size_t)W + pf, 0, 1);
            }
        }
    }

    const float om = omega[i];

    // k = omega*sqrt(e)*sqrt(m); p = sqrt(e)*rcp(sqrt(m)); invp = rcp(p).
    float kk[6], pp[6], ip[6];
#pragma unroll
    for (int l = 0; l < 6; ++l) {
        const float e = eps[l * (size_t)W + i];
        const float m = mu[l * (size_t)W + i];
        const float se = hsqrt(e);
        const float sm = hsqrt(m);
        kk[l] = om * se * sm;
        pp[l] = se * hrcp(sm);
        ip[l] = sm * hrcp(se);
    }

    // Chain of shell matrices for layers 1..4, tracked as 4 reals:
    // M = [[A, iB],[iC, D]].
    float A = 1.0f, B = 0.0f, C = 0.0f, D = 0.0f;
#pragma unroll
    for (int l = 1; l <= 4; ++l) {
        float a, b, c, d;
        shell_matrix(kk[l], pp[l], ip[l], rho[2 * l], rho[2 * l + 1],
                     a, b, c, d);
        const float nA = fmaf(A, a, -B * c);
        const float nB = fmaf(A, b, B * d);
        const float nC = fmaf(C, a, D * c);
        const float nD = fmaf(D, d, -C * b);
        A = nA; B = nB; C = nC; D = nD;
    }

    // Boundary Hankel ratios C(z) = H0'(z)/H0(z); c1 = conj(c2) for real z.
    // c2 = -[(j1 j0 + y1 y0) + i (j1 y0 - y1 j0)] / (j0^2 + y0^2)
    float j0v, j1v, y0v, y1v;

    bessel4(kk[0] * rho[1], j0v, j1v, y0v, y1v);  // inner medium, outer radius
    float inv = hrcp(fmaf(j0v, j0v, y0v * y0v));
    const float c02r = -fmaf(j1v, j0v, y1v * y0v) * inv;
    const float c02i = -fmaf(j1v, y0v, -y1v * j0v) * inv;

    bessel4(kk[5] * rho[10], j0v, j1v, y0v, y1v); // outer medium, inner radius
    inv = hrcp(fmaf(j0v, j0v, y0v * y0v));
    const float c12r = -fmaf(j1v, j0v, y1v * y0v) * inv;
    const float c12i = -fmaf(j1v, y0v, -y1v * j0v) * inv;

    const float p0 = pp[0];
    const float p1 = pp[5];
    const float g0r = p0 * c02r;     // p0 * c0_2
    const float g0i = p0 * c02i;
    const float g1r = p1 * c12r;     // p1 * c1_2
    const float g1i = p1 * c12i;

    // num ~ i * [ (C + p0 c02 A) - p1 c12 (D - p0 c02 B) ]
    const float t1r = fmaf(g0r, A, C);
    const float t1i = g0i * A;
    const float t2r = fmaf(-g0r, B, D);
    const float t2i = -g0i * B;
    const float nr = t1r - fmaf(g1r, t2r, -g1i * t2i);
    const float ni = t1i - fmaf(g1r, t2i, g1i * t2r);

    // den ~ -i * [ (p0 c01 A + C) + p1 c12 (p0 c01 B - D) ], c01 = conj(c02)
    const float u1r = fmaf(g0r, A, C);
    const float u1i = -g0i * A;
    const float u2r = fmaf(g0r, B, -D);
    const float u2i = -g0i * B;
    const float dr = u1r + fmaf(g1r, u2r, -g1i * u2i);
    const float di = u1i + fmaf(g1r, u2i, g1i * u2r);

    // |num|^2 / |den|^2 (the dropped +/- i factors cancel in magnitude)
    const float R = fmaf(nr, nr, ni * ni) * hrcp(fmaf(dr, dr, di * di));
    out[i] = R;
}

extern "C" void kernel_launch(void* const* d_in, const int* in_sizes, int n_in,
                              void* d_out, int out_size, void* d_ws,
                              size_t ws_size, hipStream_t stream) {
    (void)n_in; (void)d_ws; (void)ws_size; (void)out_size;
    const float* omega = (const float*)d_in[0];
    const float* eps   = (const float*)d_in[1];
    const float* mu    = (const float*)d_in[2];
    const float* rho   = (const float*)d_in[3];
    float* out = (float*)d_out;
    const int W = in_sizes[0];
    const int block = 256;
    const int grid = (W + block - 1) / block;
    AnnularPhotonicTransferMatrix_39075612459059_kernel<<<grid, block, 0, stream>>>(
        omega, eps, mu, rho, out, W);
}